// GraphTransformerAttention_3547642986694
// MI455X (gfx1250) — compile-verified
//
#include <hip/hip_runtime.h>

// ---------------------------------------------------------------------------
// GraphTransformerAttention on MI455X (gfx1250), wave32 + fp32 WMMA +
// async-to-LDS staging. Block-diagonal structure: the -1e9 cross-graph mask
// makes softmax exactly per-graph (exp underflows to 0), so attention is 32
// independent 64x64 (x8 heads) blocks instead of a 2048x2048 score tensor.
// ---------------------------------------------------------------------------

typedef __attribute__((ext_vector_type(2))) float v2f;
typedef __attribute__((ext_vector_type(8))) float v8f;

#define N_NODES   2048
#define G_GRAPHS  32
#define NPG       64
#define E_EDGES   16384
#define HIDDEN    256
#define N_HEADS   8
#define HEAD_DIM  32
#define EDGE_FEAT 16

__device__ __forceinline__ v8f wmma_f32(v2f a, v2f b, v8f c) {
  // V_WMMA_F32_16X16X4_F32 : D = A(16x4) * B(4x16) + C(16x16), fp32 throughout
  return __builtin_amdgcn_wmma_f32_16x16x4_f32(
      /*neg_a=*/false, a, /*neg_b=*/false, b,
      /*c_mod=*/(short)0, c, /*reuse_a=*/false, /*reuse_b=*/false);
}

// Async DMA: global memory -> LDS, no VGPR round-trip (ASYNCcnt-tracked).
__device__ __forceinline__ void async_ld_b128(unsigned lds_off, const float* g) {
  asm volatile("global_load_async_to_lds_b128 %0, %1, off"
               :: "v"(lds_off), "v"(g) : "memory");
}
__device__ __forceinline__ void wait_async0() {
  asm volatile("s_wait_asynccnt 0x0" ::: "memory");
}

// ---------------------------------------------------------------------------
__global__ void zero_kernel(float* __restrict__ p, int n) {
  int i = blockIdx.x * blockDim.x + threadIdx.x;
  if (i < n) p[i] = 0.0f;
}

// Dense per-graph edge bias: ebias[recv_global][sender_local][head]
__global__ void edge_bias_kernel(const float* __restrict__ edges,
                                 const float* __restrict__ We,
                                 const float* __restrict__ be,
                                 const int* __restrict__ senders,
                                 const int* __restrict__ receivers,
                                 float* __restrict__ ebias) {
  int e = blockIdx.x * blockDim.x + threadIdx.x;
  if (e >= E_EDGES) return;
  float ef[EDGE_FEAT];
#pragma unroll
  for (int f = 0; f < EDGE_FEAT; ++f) ef[f] = edges[(size_t)e * EDGE_FEAT + f];
  int r = receivers[e];
  int s = senders[e];
  int sl = s & (NPG - 1);  // sender local to its graph (same graph as r)
  float* dst = ebias + ((size_t)r * NPG + sl) * N_HEADS;
#pragma unroll
  for (int h = 0; h < N_HEADS; ++h) {
    float acc = be[h];
#pragma unroll
    for (int f = 0; f < EDGE_FEAT; ++f) acc += ef[f] * We[f * N_HEADS + h];
    dst[h] = acc;  // (r,s) pairs unique within a graph -> plain store == .set
  }
}

// ---------------------------------------------------------------------------
// C[M x Ncols] = A[M x K] @ W[K x Ncols] + bias ; one wave per 16x16 tile.
// f32 WMMA operand layout (ISA 7.12.2):
//   A 16x4 : lanes 0-15 rows M=lane,  v0=K0 v1=K1 ; lanes 16-31 v0=K2 v1=K3
//   B 4x16 : lanes 0-15 cols N=lane,  v0=K0 v1=K1 ; lanes 16-31 v0=K2 v1=K3
//   C/D    : vgpr r -> (M = r + 8*half, N = lane&15)
__global__ void gemm_wmma(const float* __restrict__ A,
                          const float* __restrict__ W,
                          const float* __restrict__ bias,
                          float* __restrict__ C,
                          int K, int Ncols) {
  int lane = threadIdx.x & 31;
  int l15  = lane & 15;
  int half = lane >> 4;
  int n0 = blockIdx.x * 16;
  int m0 = blockIdx.y * 16;
  int m = m0 + l15;
  int n = n0 + l15;
  v8f c = {};
  for (int k0 = 0; k0 < K; k0 += 4) {
    int ka = k0 + half * 2;
    if (k0 + 8 < K)  // speculative prefetch of the W panel two steps ahead
      __builtin_prefetch(&W[(size_t)(ka + 8) * Ncols + n], 0, 3);
    v2f a, b;
    a.x = A[(size_t)m * K + ka];
    a.y = A[(size_t)m * K + ka + 1];
    b.x = W[(size_t)ka * Ncols + n];
    b.y = W[(size_t)(ka + 1) * Ncols + n];
    c = wmma_f32(a, b, c);
  }
  float bb = bias[n];
#pragma unroll
  for (int r = 0; r < 8; ++r) {
    int row = m0 + r + half * 8;
    C[(size_t)row * Ncols + n] = c[r] + bb;
  }
}

// ---------------------------------------------------------------------------
// One block = one (graph, head). 8 waves / 256 threads.
// S = (q @ k^T) * scale + edge_bias ; P = softmax_rows(S) ; ctx = P @ v
__global__ void attn_kernel(const float* __restrict__ qb,
                            const float* __restrict__ kb,
                            const float* __restrict__ vb,
                            const float* __restrict__ ebias,
                            float* __restrict__ ctx) {
  // +4 pad: 144B row stride keeps 16B alignment for B128 async stores while
  // staying bank-conflict-free for column walks (36*16 == 9*64).
  __shared__ float sq[NPG][HEAD_DIM + 4];
  __shared__ float sk[NPG][HEAD_DIM + 4];
  __shared__ float sv[NPG][HEAD_DIM + 4];
  __shared__ float sS[NPG][NPG + 1];

  int g = blockIdx.x >> 3;   // / N_HEADS
  int h = blockIdx.x & 7;
  int tid = threadIdx.x;

  // Stage q/k/v head-slices into LDS via the CDNA5 async engine (B128).
  for (int idx = tid; idx < NPG * (HEAD_DIM / 4); idx += 256) {
    int row  = idx >> 3;        // / 8 float4s per row
    int col  = (idx & 7) * 4;
    size_t goff = (size_t)(g * NPG + row) * HIDDEN + h * HEAD_DIM + col;
    async_ld_b128((unsigned)(uintptr_t)&sq[row][col], qb + goff);
    async_ld_b128((unsigned)(uintptr_t)&sk[row][col], kb + goff);
    async_ld_b128((unsigned)(uintptr_t)&sv[row][col], vb + goff);
  }
  wait_async0();        // ASYNCcnt==0: this wave's LDS writes are visible
  __syncthreads();      // cross-wave visibility

  int wave = tid >> 5;
  int lane = tid & 31;
  int l15  = lane & 15;
  int half = lane >> 4;
  const float scale = 0.17677669529663687f;  // HEAD_DIM^-0.5

  // ---- scores: 64x64 = 16 tiles of 16x16, 2 tiles per wave, K = 32 ----
#pragma unroll
  for (int tt = 0; tt < 2; ++tt) {
    int t = wave * 2 + tt;
    int ti = t >> 2, tj = t & 3;
    v8f c = {};
#pragma unroll
    for (int k0 = 0; k0 < HEAD_DIM; k0 += 4) {
      int ka = k0 + half * 2;
      v2f a, b;
      a.x = sq[ti * 16 + l15][ka];
      a.y = sq[ti * 16 + l15][ka + 1];
      b.x = sk[tj * 16 + l15][ka];   // B = k^T : column n of B is row n of k
      b.y = sk[tj * 16 + l15][ka + 1];
      c = wmma_f32(a, b, c);
    }
#pragma unroll
    for (int r = 0; r < 8; ++r) {
      int row = ti * 16 + r + half * 8;
      int col = tj * 16 + l15;
      float bias = ebias[((size_t)(g * NPG + row) * NPG + col) * N_HEADS + h];
      sS[row][col] = c[r] * scale + bias;
    }
  }
  __syncthreads();

  // ---- row softmax (waves 0-1 active; wave-uniform branch) ----
  if (tid < NPG) {
    float mx = -1e30f;
    for (int j = 0; j < NPG; ++j) mx = fmaxf(mx, sS[tid][j]);
    float sum = 0.0f;
    for (int j = 0; j < NPG; ++j) {
      float ev = expf(sS[tid][j] - mx);
      sS[tid][j] = ev;
      sum += ev;
    }
    float inv = 1.0f / sum;
    for (int j = 0; j < NPG; ++j) sS[tid][j] *= inv;
  }
  __syncthreads();

  // ---- ctx = P(64x64) @ v(64x32) : 8 tiles, one per wave, K = 64 ----
  {
    int ti = wave >> 1, tj = wave & 1;
    v8f c = {};
#pragma unroll
    for (int k0 = 0; k0 < NPG; k0 += 4) {
      int ka = k0 + half * 2;
      v2f a, b;
      a.x = sS[ti * 16 + l15][ka];
      a.y = sS[ti * 16 + l15][ka + 1];
      b.x = sv[ka][tj * 16 + l15];
      b.y = sv[ka + 1][tj * 16 + l15];
      c = wmma_f32(a, b, c);
    }
#pragma unroll
    for (int r = 0; r < 8; ++r) {
      int row = ti * 16 + r + half * 8;
      int col = tj * 16 + l15;
      ctx[(size_t)(g * NPG + row) * HIDDEN + h * HEAD_DIM + col] = c[r];
    }
  }
}

// ---------------------------------------------------------------------------
extern "C" void kernel_launch(void* const* d_in, const int* in_sizes, int n_in,
                              void* d_out, int out_size, void* d_ws, size_t ws_size,
                              hipStream_t stream) {
  const float* nodes     = (const float*)d_in[0];
  const float* edges     = (const float*)d_in[1];
  // d_in[2] = n_node (always NPG, unused)
  const int*   senders   = (const int*)d_in[3];
  const int*   receivers = (const int*)d_in[4];
  const float* Wq = (const float*)d_in[5];
  const float* bq = (const float*)d_in[6];
  const float* Wk = (const float*)d_in[7];
  const float* bk = (const float*)d_in[8];
  const float* Wv = (const float*)d_in[9];
  const float* bv = (const float*)d_in[10];
  const float* Wo = (const float*)d_in[11];
  const float* bo = (const float*)d_in[12];
  const float* We = (const float*)d_in[13];
  const float* be = (const float*)d_in[14];
  float* out = (float*)d_out;

  // Workspace layout (floats): q | k | v | ebias | ctx  = 12 MB total
  float* ws    = (float*)d_ws;
  float* qb    = ws;
  float* kb    = qb + (size_t)N_NODES * HIDDEN;
  float* vb    = kb + (size_t)N_NODES * HIDDEN;
  float* ebias = vb + (size_t)N_NODES * HIDDEN;
  float* ctx   = ebias + (size_t)N_NODES * NPG * N_HEADS;

  int ebias_n = N_NODES * NPG * N_HEADS;
  zero_kernel<<<(ebias_n + 255) / 256, 256, 0, stream>>>(ebias, ebias_n);
  edge_bias_kernel<<<(E_EDGES + 255) / 256, 256, 0, stream>>>(
      edges, We, be, senders, receivers, ebias);

  dim3 ggrid(HIDDEN / 16, N_NODES / 16);  // one wave per 16x16 output tile
  gemm_wmma<<<ggrid, 32, 0, stream>>>(nodes, Wq, bq, qb, HIDDEN, HIDDEN);
  gemm_wmma<<<ggrid, 32, 0, stream>>>(nodes, Wk, bk, kb, HIDDEN, HIDDEN);
  gemm_wmma<<<ggrid, 32, 0, stream>>>(nodes, Wv, bv, vb, HIDDEN, HIDDEN);

  attn_kernel<<<G_GRAPHS * N_HEADS, 256, 0, stream>>>(qb, kb, vb, ebias, ctx);

  gemm_wmma<<<ggrid, 32, 0, stream>>>(ctx, Wo, bo, out, HIDDEN, HIDDEN);
}